// Random_frequency_replacing_4707284156637
// MI455X (gfx1250) — compile-verified
//
#include <hip/hip_runtime.h>
#include <math.h>

// ---------------------------------------------------------------------------
// out = clean + Re( IFFT2( M_b * FFT2(noisy-clean) ) )
// Tensor-core (WMMA f32) 512-pt FFT: 512 = 16 x 32 Cooley-Tukey.
//   inner : D1[n1,k2] = sum_n2 X[n1,n2] * W32[n2,k2]
//   twiddle: Atw = D1 * W512^{n1*k2}
//   outer : E[k2,k1] = sum_n1 Atw^T[k2,n1] * W16[n1,k1]  -> x[32*k1+k2]
// One wave == one transform; DFT/twiddle constants in per-block LDS tables;
// each wave runs RPW transforms. ifft(X)=conj(fft(conj X))/N and the
// inter-pass conjugations cancel -> every FFT is the same forward transform.
// Passes 2/3 double-buffer their input with GLOBAL_LOAD_ASYNC_TO_LDS_B64
// (ASYNCcnt): the async engine streams the next column/row into LDS while
// the wave's WMMAs chew on the current one.
// ---------------------------------------------------------------------------

#define NI   48            // B*C images
#define WPB  4             // waves per block
#define TPB  (WPB * 32)
#define RPW  8             // transforms per wave (amortizes LDS tables)
#define LDSN 528           // 512 complex + pad (idx k -> k + k/32): no 16-way conflicts

typedef __attribute__((ext_vector_type(2))) float v2f;
typedef __attribute__((ext_vector_type(8))) float v8f;

struct FftTab {
  float4 tin[16][32];   // stage-inner W32 B-frags: [(t*2+s2)][lane] = {c0,c1,s0,s1}
  float4 tout[4][32];   // stage-outer W16 B-frags: [t][lane]
  float2 ttw[16][32];   // 512-pt twiddles:        [(s2*8+r)][lane] = {cos,sin}
};

__device__ __forceinline__ int lix(int k) { return k + (k >> 5); }

__device__ __forceinline__ v8f wmma4(v2f a, v2f b, v8f c) {
  // D(16x16,f32) = A(16x4,f32) * B(4x16,f32) + C
  return __builtin_amdgcn_wmma_f32_16x16x4_f32(false, a, false, b, (short)0, c,
                                               false, false);
}

__device__ __forceinline__ void wavefence() {
  __builtin_amdgcn_wave_barrier();
  asm volatile("s_wait_dscnt 0" ::: "memory");
  __builtin_amdgcn_wave_barrier();
}

// --- async global->LDS staging (CDNA5 TDM-side path, tracked by ASYNCcnt) ---
// LDS byte offset == low 32 bits of the flat shared-aperture address.
__device__ __forceinline__ void async_ld64(const float2* g, const float2* l) {
  unsigned loff = (unsigned)(uintptr_t)l;
  unsigned long long ga = (unsigned long long)(uintptr_t)g;
  asm volatile("global_load_async_to_lds_b64 %0, %1, off"
               :: "v"(loff), "v"(ga) : "memory");
}
// 16 b64 loads = one 512-pt column (stride 512 complex) into padded LDS line
__device__ __forceinline__ void prefetch_col(const float2* zcol, float2* dst, int lane) {
#pragma unroll
  for (int i = 0; i < 16; ++i) {
    int u = lane + 32 * i;
    async_ld64(zcol + ((size_t)u << 9), dst + lix(u));
  }
}
// 16 b64 loads = one contiguous 512-pt row into padded LDS line
__device__ __forceinline__ void prefetch_row(const float2* zrow, float2* dst, int lane) {
#pragma unroll
  for (int i = 0; i < 16; ++i) {
    int k = lane + 32 * i;
    async_ld64(zrow + k, dst + lix(k));
  }
}
__device__ __forceinline__ void wait_async16() {
  asm volatile("s_wait_asynccnt 0x10" ::: "memory");  // newest 16 may remain
}
__device__ __forceinline__ void wait_async0() {
  asm volatile("s_wait_asynccnt 0x0" ::: "memory");
}
__device__ __forceinline__ void drain_lds_reads() {
  asm volatile("s_wait_dscnt 0x0" ::: "memory");      // before async overwrites a buffer
}

// Build forward-FFT constant tables (one wave writes, block-wide visibility).
__device__ void build_tables(FftTab* T) {
  if (threadIdx.x < 32) {
    const int lane = threadIdx.x, m = lane & 15, h = lane >> 4;
    const float PI2 = 6.28318530717958647692f;
#pragma unroll
    for (int t = 0; t < 8; ++t) {
      int k0 = 4 * t + 2 * h;  // B row (n2); component .y is row k0+1
#pragma unroll
      for (int s2 = 0; s2 < 2; ++s2) {
        int col = 16 * s2 + m;  // k2
        float a0 = -(PI2 / 32.f) * (float)((k0 * col) & 31);
        float a1 = -(PI2 / 32.f) * (float)(((k0 + 1) * col) & 31);
        float s0, c0, s1, c1;
        __sincosf(a0, &s0, &c0);
        __sincosf(a1, &s1, &c1);
        T->tin[t * 2 + s2][lane] = make_float4(c0, c1, s0, s1);
      }
    }
#pragma unroll
    for (int t = 0; t < 4; ++t) {
      int k0 = 4 * t + 2 * h;  // B row (n1); col = m (k1)
      float a0 = -(PI2 / 16.f) * (float)((k0 * m) & 15);
      float a1 = -(PI2 / 16.f) * (float)(((k0 + 1) * m) & 15);
      float s0, c0, s1, c1;
      __sincosf(a0, &s0, &c0);
      __sincosf(a1, &s1, &c1);
      T->tout[t][lane] = make_float4(c0, c1, s0, s1);
    }
#pragma unroll
    for (int s2 = 0; s2 < 2; ++s2) {
      int k2 = 16 * s2 + m;
#pragma unroll
      for (int r = 0; r < 8; ++r) {
        int n1 = r + 8 * h;
        float ang = -(PI2 / 512.f) * (float)(n1 * k2);
        float sn, cs;
        __sincosf(ang, &sn, &cs);
        T->ttw[s2 * 8 + r][lane] = make_float2(cs, sn);
      }
    }
  }
  __syncthreads();
}

// In-place forward 512-pt FFT on wave-private LDS line sb (padded idx lix(k)).
template <bool REALIN>
__device__ void fft512(float2* sb, const FftTab* T) {
  const int lane = threadIdx.x & 31;
  const int m = lane & 15;
  const int h = lane >> 4;

  wavefence();

  // ---- load A-frags: X[n1,n2] = x[n1 + 16*n2]; chunk t: K = 4t + j + 2h
  v2f Ar[8], Ai[8];
#pragma unroll
  for (int t = 0; t < 8; ++t) {
    int n2 = 4 * t + 2 * h;
    float2 e0 = sb[lix(m + 16 * n2)];
    float2 e1 = sb[lix(m + 16 * (n2 + 1))];
    Ar[t] = {e0.x, e1.x};
    Ai[t] = {e0.y, e1.y};
  }

  // ---- stage inner: D1 = X * W32 (two 16-wide k2 tiles)
  v8f Cr[2], Ci[2];
#pragma unroll
  for (int s2 = 0; s2 < 2; ++s2) {
    v8f cr = {0.f, 0.f, 0.f, 0.f, 0.f, 0.f, 0.f, 0.f};
    v8f ci = {0.f, 0.f, 0.f, 0.f, 0.f, 0.f, 0.f, 0.f};
#pragma unroll
    for (int t = 0; t < 8; ++t) {
      float4 wv = T->tin[t * 2 + s2][lane];
      v2f br = {wv.x, wv.y}, bi = {wv.z, wv.w};
      cr = wmma4(Ar[t], br, cr);
      ci = wmma4(Ar[t], bi, ci);
      if (!REALIN) {
        v2f nbi = {-wv.z, -wv.w};
        cr = wmma4(Ai[t], nbi, cr);
        ci = wmma4(Ai[t], br, ci);
      }
    }
    Cr[s2] = cr;
    Ci[s2] = ci;
  }

  // ---- twiddle: * W512^{n1*k2} (C-layout: VGPR r, half h -> n1=r+8h, k2=16s2+m)
#pragma unroll
  for (int s2 = 0; s2 < 2; ++s2) {
#pragma unroll
    for (int r = 0; r < 8; ++r) {
      float2 tw = T->ttw[s2 * 8 + r][lane];
      float xr = Cr[s2][r], xi = Ci[s2][r];
      Cr[s2][r] = xr * tw.x - xi * tw.y;
      Ci[s2][r] = xr * tw.y + xi * tw.x;
    }
  }

  // ---- relayout into A'-frags of Atw^T (half-wave swap via shfl_xor 16)
  v2f Apr[2][4], Api[2][4];
#pragma unroll
  for (int s2 = 0; s2 < 2; ++s2) {
#pragma unroll
    for (int t = 0; t < 4; ++t) {
      float vr[2], vi[2];
#pragma unroll
      for (int j = 0; j < 2; ++j) {
        float xr, xi;
        if (t < 2) {  // rows 4t+j (half0) / 4t+2+j (half1), all < 8
          float tr = __shfl_xor(Cr[s2][4 * t + 2 + j], 16, 32);
          float ti = __shfl_xor(Ci[s2][4 * t + 2 + j], 16, 32);
          xr = h ? tr : Cr[s2][4 * t + j];
          xi = h ? ti : Ci[s2][4 * t + j];
        } else {      // rows >= 8 live in half1: VGPR index - 8
          float tr = __shfl_xor(Cr[s2][4 * t + j - 8], 16, 32);
          float ti = __shfl_xor(Ci[s2][4 * t + j - 8], 16, 32);
          xr = h ? Cr[s2][4 * t + 2 + j - 8] : tr;
          xi = h ? Ci[s2][4 * t + 2 + j - 8] : ti;
        }
        vr[j] = xr;
        vi[j] = xi;
      }
      Apr[s2][t] = {vr[0], vr[1]};
      Api[s2][t] = {vi[0], vi[1]};
    }
  }

  // ---- stage outer: E = Atw^T * W16 ; store E[k2,k1] -> x[32*k1 + k2]
  v2f obr[4], obi[4], onbi[4];
#pragma unroll
  for (int t = 0; t < 4; ++t) {
    float4 wv = T->tout[t][lane];
    obr[t] = {wv.x, wv.y};
    obi[t] = {wv.z, wv.w};
    onbi[t] = {-wv.z, -wv.w};
  }
#pragma unroll
  for (int s2 = 0; s2 < 2; ++s2) {
    v8f er = {0.f, 0.f, 0.f, 0.f, 0.f, 0.f, 0.f, 0.f};
    v8f ei = {0.f, 0.f, 0.f, 0.f, 0.f, 0.f, 0.f, 0.f};
#pragma unroll
    for (int t = 0; t < 4; ++t) {
      er = wmma4(Apr[s2][t], obr[t], er);
      er = wmma4(Api[s2][t], onbi[t], er);
      ei = wmma4(Apr[s2][t], obi[t], ei);
      ei = wmma4(Api[s2][t], obr[t], ei);
    }
#pragma unroll
    for (int r = 0; r < 8; ++r) {
      int k = 32 * m + 16 * s2 + r + 8 * h;
      sb[lix(k)] = make_float2(er[r], ei[r]);
    }
  }

  wavefence();
}

// ---------------- pass 1: rows, forward FFT of d = noisy - clean ------------
__global__ __launch_bounds__(TPB) void k_rows_fwd(const float* __restrict__ noisy,
                                                  const float* __restrict__ clean,
                                                  float2* __restrict__ Z) {
  __shared__ FftTab tab;
  __shared__ float2 sbuf[WPB][LDSN];
  build_tables(&tab);
  const int w = threadIdx.x >> 5, lane = threadIdx.x & 31;
  float2* sb = sbuf[w];
  const int gid0 = (blockIdx.x * WPB + w) * RPW;
  for (int rr = 0; rr < RPW; ++rr) {
    const int gid = gid0 + rr;  // [0, NI*512)
    const size_t base = ((size_t)(gid >> 9) << 18) + ((size_t)(gid & 511) << 9);
    const float* np = noisy + base;
    const float* cp = clean + base;
#pragma unroll
    for (int i = 0; i < 16; ++i) {
      int v = lane + 32 * i;
      sb[lix(v)] = make_float2(np[v] - cp[v], 0.f);
    }
    fft512<true>(sb, &tab);
    float2* zp = Z + base;
#pragma unroll
    for (int i = 0; i < 16; ++i) {
      int v = lane + 32 * i;
      zp[v] = sb[lix(v)];
    }
  }
}

// -------- pass 2: columns, fwd FFT -> conj * mask/N^2 -> fwd FFT ------------
// Input columns double-buffered via async global->LDS loads.
__global__ __launch_bounds__(TPB) void k_cols(const float* __restrict__ vset,
                                              float2* __restrict__ Z) {
  __shared__ FftTab tab;
  __shared__ float2 sbuf[WPB][2][LDSN];
  build_tables(&tab);
  const int w = threadIdx.x >> 5, lane = threadIdx.x & 31;
  const int gid0 = (blockIdx.x * WPB + w) * RPW;

  const float2* Zc = (const float2*)Z;
  prefetch_col(Zc + (((size_t)(gid0 >> 9)) << 18) + (gid0 & 511), sbuf[w][0], lane);

  for (int rr = 0; rr < RPW; ++rr) {
    const int gid = gid0 + rr;
    const int img = gid >> 9, v = gid & 511;
    float2* sb = sbuf[w][rr & 1];
    float2* zp = Z + ((size_t)img << 18) + v;

    if (rr + 1 < RPW) {  // overlap next column's gather with this FFT work
      const int g2 = gid + 1;
      drain_lds_reads();
      prefetch_col(Zc + (((size_t)(g2 >> 9)) << 18) + (g2 & 511),
                   sbuf[w][(rr + 1) & 1], lane);
      wait_async16();    // in-order: current column's 16 loads have landed
    } else {
      wait_async0();
    }

    fft512<false>(sb, &tab);

    // mask: batch fftshift -> value_set[(b+8)%16]; unshifted freq coord
    // a(t) = t<256 ? t : t-512; band = floor(dist/step), 0 beyond band 99.
    // Conj folded here (its partner in pass 3 cancels); 1/N^2 in the scale.
    const int b = img / 3;
    const float* vs = vset + ((b + 8) & 15) * 100;
    const float av = (float)(v < 256 ? v : v - 512);
    const float invstep = 100.f / 362.03867196751236f;  // 1/(max_radius/100)
#pragma unroll
    for (int i = 0; i < 16; ++i) {
      int u = lane + 32 * i;
      float au = (float)(u < 256 ? u : u - 512);
      float d = sqrtf(au * au + av * av);
      int idx = (int)(d * invstep);
      float mv = (idx < 100) ? vs[idx] : 0.f;
      float sc = mv * (1.f / 262144.f);
      float2 z = sb[lix(u)];
      sb[lix(u)] = make_float2(z.x * sc, -(z.y * sc));  // conj folded in
    }
    fft512<false>(sb, &tab);
#pragma unroll
    for (int i = 0; i < 16; ++i) {
      int u = lane + 32 * i;
      zp[(size_t)u << 9] = sb[lix(u)];
    }
  }
}

// -------- pass 3: rows, fwd FFT (== conj of ifft), out = clean + Re(.) ------
// Input rows double-buffered via async global->LDS loads.
__global__ __launch_bounds__(TPB) void k_rows_inv(const float* __restrict__ clean,
                                                  float2* __restrict__ Z,
                                                  float* __restrict__ out) {
  __shared__ FftTab tab;
  __shared__ float2 sbuf[WPB][2][LDSN];
  build_tables(&tab);
  const int w = threadIdx.x >> 5, lane = threadIdx.x & 31;
  const int gid0 = (blockIdx.x * WPB + w) * RPW;

  const float2* Zc = (const float2*)Z;
  prefetch_row(Zc + ((size_t)gid0 << 9), sbuf[w][0], lane);

  for (int rr = 0; rr < RPW; ++rr) {
    const int gid = gid0 + rr;
    const size_t base = ((size_t)(gid >> 9) << 18) + ((size_t)(gid & 511) << 9);
    float2* sb = sbuf[w][rr & 1];

    if (rr + 1 < RPW) {
      drain_lds_reads();
      prefetch_row(Zc + ((size_t)(gid + 1) << 9), sbuf[w][(rr + 1) & 1], lane);
      wait_async16();
    } else {
      wait_async0();
    }

    fft512<false>(sb, &tab);
    const float* cp = clean + base;
    float* op = out + base;
#pragma unroll
    for (int i = 0; i < 16; ++i) {
      int v = lane + 32 * i;
      op[v] = cp[v] + sb[lix(v)].x;  // Re(conj(q)) == Re(q)
    }
  }
}

extern "C" void kernel_launch(void* const* d_in, const int* in_sizes, int n_in,
                              void* d_out, int out_size, void* d_ws, size_t ws_size,
                              hipStream_t stream) {
  const float* clean = (const float*)d_in[0];
  const float* noisy = (const float*)d_in[1];
  const float* vset  = (const float*)d_in[2];
  float* out = (float*)d_out;
  float2* Z = (float2*)d_ws;  // 48*512*512 complex64 = 100.7 MB (fits in 192MB L2)

  const int blocks = (NI * 512) / (WPB * RPW);  // 768 blocks, 1 wave = RPW FFTs
  k_rows_fwd<<<blocks, TPB, 0, stream>>>(noisy, clean, Z);
  k_cols<<<blocks, TPB, 0, stream>>>(vset, Z);
  k_rows_inv<<<blocks, TPB, 0, stream>>>(clean, Z, out);
}